// Transformer_2D_18700287607490
// MI455X (gfx1250) — compile-verified
//
#include <hip/hip_runtime.h>
#include <stdint.h>

// Problem shape from the reference: src [N,C,H,W] = [16,8,512,512] f32,
// flow [N,2,H,W] f32 (plane 0 = row/y displacement, plane 1 = col/x).
#define N_  16
#define C_  8
#define H_  512
#define W_  512
#define HW_ (H_ * W_)
#define BLK 256

typedef __attribute__((address_space(1))) int gint_t;   // global int
typedef __attribute__((address_space(3))) int lint_t;   // LDS int

__global__ __launch_bounds__(BLK) void flowwarp_bilinear_kernel(
    const float* __restrict__ src,
    const float* __restrict__ flow,
    float* __restrict__ out)
{
    __shared__ float sflow[2 * BLK];

    const int tid = threadIdx.x;
    const int t   = blockIdx.x * BLK + tid;       // exact grid: no bounds check
    const int w   = t & (W_ - 1);
    const int h   = (t >> 9) & (H_ - 1);
    const int n   = t >> 18;

    // ---- stage flow_y / flow_x through the async LDS-copy engine (ASYNCcnt) ----
    const float* fyp = flow + (n * 2) * HW_ + h * W_ + w;   // flow[n,0,h,w]
    const float* fxp = fyp + HW_;                            // flow[n,1,h,w]

#if __has_builtin(__builtin_amdgcn_global_load_async_to_lds_b32)
    __builtin_amdgcn_global_load_async_to_lds_b32(
        (gint_t*)(uintptr_t)fyp,
        (lint_t*)(uintptr_t)&sflow[tid], 0, 0);
    __builtin_amdgcn_global_load_async_to_lds_b32(
        (gint_t*)(uintptr_t)fxp,
        (lint_t*)(uintptr_t)&sflow[tid + BLK], 0, 0);
#else
    {
        uint32_t l0 = (uint32_t)(uintptr_t)&sflow[tid];
        uint32_t l1 = (uint32_t)(uintptr_t)&sflow[tid + BLK];
        uint64_t g0 = (uint64_t)(uintptr_t)fyp;
        uint64_t g1 = (uint64_t)(uintptr_t)fxp;
        asm volatile("global_load_async_to_lds_b32 %0, %1, off"
                     :: "v"(l0), "v"(g0) : "memory");
        asm volatile("global_load_async_to_lds_b32 %0, %1, off"
                     :: "v"(l1), "v"(g1) : "memory");
    }
#endif

    // Overlap window: per-pixel bases that don't depend on flow values.
    const int plane0 = n * C_ * HW_;
    const float* p = src + plane0;                 // channel-0 src plane
    float*       q = out + plane0 + h * W_ + w;    // channel-0 out element

#if __has_builtin(__builtin_amdgcn_s_wait_asynccnt)
    __builtin_amdgcn_s_wait_asynccnt(0);
#else
    asm volatile("s_wait_asynccnt 0" ::: "memory");
#endif
    asm volatile("" ::: "memory");                 // order LDS reads after the wait

    const float fy = sflow[tid];
    const float fx = sflow[tid + BLK];

    // Replicate the reference's normalize -> denormalize chain in f32.
    const float ny = 2.0f * (((float)h + fy) / (float)(H_ - 1) - 0.5f);
    const float nx = 2.0f * (((float)w + fx) / (float)(W_ - 1) - 0.5f);
    const float iy = (ny + 1.0f) * 0.5f * (float)(H_ - 1);
    const float ix = (nx + 1.0f) * 0.5f * (float)(W_ - 1);

    const float x0f = floorf(ix);
    const float y0f = floorf(iy);

    // Bilinear weights, same corner algebra as the reference.
    const float wnw = (x0f + 1.0f - ix) * (y0f + 1.0f - iy);
    const float wne = (ix - x0f)        * (y0f + 1.0f - iy);
    const float wsw = (x0f + 1.0f - ix) * (iy - y0f);
    const float wse = (ix - x0f)        * (iy - y0f);

    // Clamp in float (safe for huge/neg values) then convert.
    const int x0 = (int)fminf(fmaxf(x0f,        0.0f), (float)(W_ - 1));
    const int x1 = (int)fminf(fmaxf(x0f + 1.0f, 0.0f), (float)(W_ - 1));
    const int y0 = (int)fminf(fmaxf(y0f,        0.0f), (float)(H_ - 1));
    const int y1 = (int)fminf(fmaxf(y0f + 1.0f, 0.0f), (float)(H_ - 1));

    const int o00 = y0 * W_ + x0;
    const int o01 = y0 * W_ + x1;
    const int o10 = y1 * W_ + x0;
    const int o11 = y1 * W_ + x1;

    // 8 channels share one set of weights/offsets: 32 gathers, 8 NT stores.
#pragma unroll
    for (int c = 0; c < C_; ++c) {
        const float v00 = p[o00];
        const float v01 = p[o01];
        const float v10 = p[o10];
        const float v11 = p[o11];
        const float r = wnw * v00 + wne * v01 + wsw * v10 + wse * v11;
        // Output is 128 MB write-once: keep it out of L2 so src stays resident.
        __builtin_nontemporal_store(r, q);
        p += HW_;
        q += HW_;
    }
}

extern "C" void kernel_launch(void* const* d_in, const int* in_sizes, int n_in,
                              void* d_out, int out_size, void* d_ws, size_t ws_size,
                              hipStream_t stream) {
    (void)in_sizes; (void)n_in; (void)d_ws; (void)ws_size; (void)out_size;
    const float* src  = (const float*)d_in[0];
    const float* flow = (const float*)d_in[1];
    float*       out  = (float*)d_out;

    const int pixels = N_ * H_ * W_;              // 4,194,304
    dim3 grid(pixels / BLK), block(BLK);
    flowwarp_bilinear_kernel<<<grid, block, 0, stream>>>(src, flow, out);
}